// ProposalLayer_24421184045199
// MI455X (gfx1250) — compile-verified
//
#include <hip/hip_runtime.h>
#include <stdint.h>

#define BATCH 8
#define NANCH 261888
#define KTOP  6000
#define PROP  1000
#define NBINS 4096
#define CAP   8192
#define NMS_THR 0.7f
#define NW 188               // ceil(6000/32) mask dwords

// ---- workspace layout (bytes) ----
#define OFF_HIST  0                                   // BATCH*NBINS*4 = 131072
#define OFF_TBIN  (OFF_HIST + BATCH*NBINS*4)          // 32 B
#define OFF_CNT   (OFF_TBIN + BATCH*4)                // 32 B
#define OFF_KEYS  (OFF_CNT  + BATCH*4)                // BATCH*CAP*8 = 524288
#define OFF_BOXES (OFF_KEYS + BATCH*CAP*8)            // BATCH*KTOP*4*4 = 768000
#define OFF_AREAS (OFF_BOXES + BATCH*KTOP*16)         // BATCH*KTOP*4 = 192000

static __device__ __forceinline__ unsigned ballot32(bool p) {
#if __has_builtin(__builtin_amdgcn_ballot_w32)
  return __builtin_amdgcn_ballot_w32(p);
#else
  return (unsigned)__ballot(p);
#endif
}

__global__ void zero_kernel(unsigned* hist, unsigned* cnt) {
  int i = blockIdx.x * blockDim.x + threadIdx.x;
  if (i < BATCH * NBINS) hist[i] = 0u;
  if (i < BATCH) cnt[i] = 0u;
}

__global__ void hist_kernel(const float* __restrict__ probs,
                            unsigned* __restrict__ hist) {
  int b = blockIdx.y;
  int i = blockIdx.x * blockDim.x + threadIdx.x;
  if (i >= NANCH) return;
  float s = probs[((size_t)b * NANCH + i) * 2 + 1];
  int bin = (int)(s * (float)NBINS);
  bin = bin < 0 ? 0 : (bin > NBINS - 1 ? NBINS - 1 : bin);
  atomicAdd(&hist[b * NBINS + bin], 1u);
}

__global__ void scan_kernel(const unsigned* __restrict__ hist,
                            unsigned* __restrict__ tbin) {
  int b = threadIdx.x;
  if (b >= BATCH) return;
  unsigned cum = 0; int t = 0;
  for (int bin = NBINS - 1; bin >= 0; --bin) {
    cum += hist[b * NBINS + bin];
    if (cum >= KTOP) { t = bin; break; }
  }
  tbin[b] = (unsigned)t;
}

__global__ void compact_kernel(const float* __restrict__ probs,
                               const unsigned* __restrict__ tbin,
                               unsigned* __restrict__ cnt,
                               unsigned long long* __restrict__ keys) {
  int b = blockIdx.y;
  int i = blockIdx.x * blockDim.x + threadIdx.x;
  if (i >= NANCH) return;
  float s = probs[((size_t)b * NANCH + i) * 2 + 1];
  int bin = (int)(s * (float)NBINS);
  bin = bin < 0 ? 0 : (bin > NBINS - 1 ? NBINS - 1 : bin);
  if ((unsigned)bin >= tbin[b]) {
    unsigned pos = atomicAdd(&cnt[b], 1u);
    if (pos < CAP) {
      unsigned long long key =
          ((unsigned long long)__float_as_uint(s) << 32) |
          (unsigned long long)(~(unsigned)i);
      keys[(size_t)b * CAP + pos] = key;
    }
  }
}

__global__ __launch_bounds__(1024)
void sort_kernel(unsigned long long* __restrict__ keys,
                 const unsigned* __restrict__ cnt) {
  __shared__ unsigned long long sk[CAP];   // 64 KB
  int b = blockIdx.x;
  unsigned n = cnt[b]; if (n > CAP) n = CAP;
  unsigned long long* kb = keys + (size_t)b * CAP;
  for (int i = threadIdx.x; i < CAP; i += blockDim.x)
    sk[i] = (i < (int)n) ? kb[i] : 0ull;
  __syncthreads();
  // bitonic sort, descending
  for (unsigned ksz = 2; ksz <= CAP; ksz <<= 1) {
    for (unsigned j = ksz >> 1; j > 0; j >>= 1) {
      for (unsigned i = threadIdx.x; i < CAP; i += blockDim.x) {
        unsigned ixj = i ^ j;
        if (ixj > i) {
          unsigned long long a = sk[i], c = sk[ixj];
          bool asc = (i & ksz) != 0;
          if (asc ? (a > c) : (a < c)) { sk[i] = c; sk[ixj] = a; }
        }
      }
      __syncthreads();
    }
  }
  for (int i = threadIdx.x; i < KTOP; i += blockDim.x)
    kb[i] = sk[i];
}

__global__ void gather_kernel(const float* __restrict__ bbox,
                              const float* __restrict__ anchors,
                              const unsigned long long* __restrict__ keys,
                              float* __restrict__ boxes,
                              float* __restrict__ areas) {
  int b = blockIdx.y;
  int j = blockIdx.x * blockDim.x + threadIdx.x;
  if (j >= KTOP) return;
  unsigned long long key = keys[(size_t)b * CAP + j];
  unsigned i = ~(unsigned)key;
  float y1 = 0.f, x1 = 0.f, y2 = 0.f, x2 = 0.f;
  if (i < NANCH) {
    size_t base = ((size_t)b * NANCH + i) * 4;
    float a0 = anchors[base + 0], a1 = anchors[base + 1];
    float a2 = anchors[base + 2], a3 = anchors[base + 3];
    float d0 = bbox[base + 0] * 0.1f, d1 = bbox[base + 1] * 0.1f;
    float d2 = bbox[base + 2] * 0.2f, d3 = bbox[base + 3] * 0.2f;
    float h = a2 - a0, w = a3 - a1;
    float cy = a0 + 0.5f * h + d0 * h;
    float cx = a1 + 0.5f * w + d1 * w;
    h *= expf(d2);
    w *= expf(d3);
    y1 = cy - 0.5f * h; x1 = cx - 0.5f * w;
    y2 = y1 + h;        x2 = x1 + w;
    y1 = fminf(fmaxf(y1, 0.f), 1.f); x1 = fminf(fmaxf(x1, 0.f), 1.f);
    y2 = fminf(fmaxf(y2, 0.f), 1.f); x2 = fminf(fmaxf(x2, 0.f), 1.f);
  }
  size_t o = ((size_t)b * KTOP + j) * 4;
  boxes[o + 0] = y1; boxes[o + 1] = x1; boxes[o + 2] = y2; boxes[o + 3] = x2;
  areas[(size_t)b * KTOP + j] = (y2 - y1) * (x2 - x1);
}

__global__ __launch_bounds__(1024)
void nms_kernel(const float* __restrict__ boxes,
                const float* __restrict__ areas,
                float* __restrict__ out) {
  __shared__ float4 sb[KTOP];       // 96 KB  — CDNA5 320 KB WGP LDS
  __shared__ float sarea[KTOP];     // 24 KB
  __shared__ unsigned mask[NW];
  __shared__ int smin[2];
  __shared__ int sbase;
  int b = blockIdx.x;
  int tid = threadIdx.x;

  // CDNA5 async global -> LDS staging: boxes (6000 x b128) + areas (1500 x b128)
  {
    unsigned long long gb =
        (unsigned long long)(uintptr_t)(boxes + (size_t)b * KTOP * 4);
    unsigned lb = (unsigned)(uintptr_t)(void*)&sb[0];
    for (int c = tid; c < KTOP; c += 1024) {
      unsigned long long ga = gb + (unsigned long long)c * 16u;
      unsigned la = lb + (unsigned)c * 16u;
      asm volatile("global_load_async_to_lds_b128 %0, %1, off"
                   :: "v"(la), "v"(ga) : "memory");
    }
    unsigned long long gA =
        (unsigned long long)(uintptr_t)(areas + (size_t)b * KTOP);
    unsigned lA = (unsigned)(uintptr_t)(void*)&sarea[0];
    for (int c = tid; c < KTOP / 4; c += 1024) {
      unsigned long long ga = gA + (unsigned long long)c * 16u;
      unsigned la = lA + (unsigned)c * 16u;
      asm volatile("global_load_async_to_lds_b128 %0, %1, off"
                   :: "v"(la), "v"(ga) : "memory");
    }
    asm volatile("s_wait_asynccnt 0" ::: "memory");
  }

  for (int t = tid; t < NW; t += 1024)
    mask[t] = (t == NW - 1) ? 0x0000FFFFu : 0xFFFFFFFFu;
  if (tid == 0) { smin[0] = 0x7FFFFFFF; smin[1] = 0x7FFFFFFF; sbase = 0; }
  __syncthreads();

  for (int p = 0; p < PROP; ++p) {
    int cur = p & 1, nxt = cur ^ 1;
    // ---- find first valid index (first set bit; indices are monotone) ----
    int sbl = sbase;
    if (tid >= sbl && tid < NW) {
      unsigned m = mask[tid];
      if (m) atomicMin(&smin[cur], tid * 32 + __builtin_ctz(m));
    }
    __syncthreads();                       // B1
    int idx = smin[cur];
    bool has = idx < KTOP;
    float4 box = has ? sb[idx] : make_float4(0.f, 0.f, 0.f, 0.f);
    if (tid == 0) {
      size_t o = ((size_t)b * PROP + p) * 4;
      out[o + 0] = box.x; out[o + 1] = box.y;
      out[o + 2] = box.z; out[o + 3] = box.w;
      smin[nxt] = 0x7FFFFFFF;              // reset other slot (no extra barrier)
      if (has) sbase = idx >> 5;           // scan start is monotone
    }
    if (has) {
      float barea = sarea[idx];
      for (int base = 0; base < KTOP; base += 1024) {
        if (base + 1023 < idx) continue;   // chunk entirely below idx: dead
        int j = base + tid;
        bool keep = true;
        if (j < KTOP) {
          float4 bj = sb[j];
          float yy1 = fmaxf(box.x, bj.x);
          float xx1 = fmaxf(box.y, bj.y);
          float yy2 = fminf(box.z, bj.z);
          float xx2 = fminf(box.w, bj.w);
          float inter = fmaxf(yy2 - yy1, 0.f) * fmaxf(xx2 - xx1, 0.f);
          float uni = barea + sarea[j] - inter;
          float iou = (uni > 0.f) ? inter / uni : 0.f;
          keep = iou <= NMS_THR;
        }
        unsigned bal = ballot32(keep);     // wave32 lanes == one mask dword
        if ((tid & 31) == 0) {
          int d = (base + tid) >> 5;
          if (d < NW) mask[d] &= bal;
        }
      }
    }
    __syncthreads();                       // B2
  }
}

extern "C" void kernel_launch(void* const* d_in, const int* in_sizes, int n_in,
                              void* d_out, int out_size, void* d_ws, size_t ws_size,
                              hipStream_t stream) {
  const float* rpn_probs = (const float*)d_in[0];
  const float* rpn_bbox  = (const float*)d_in[1];
  const float* anchors   = (const float*)d_in[2];
  float* out = (float*)d_out;

  char* ws = (char*)d_ws;
  unsigned* hist = (unsigned*)(ws + OFF_HIST);
  unsigned* tbin = (unsigned*)(ws + OFF_TBIN);
  unsigned* cnt  = (unsigned*)(ws + OFF_CNT);
  unsigned long long* keys = (unsigned long long*)(ws + OFF_KEYS);
  float* boxes = (float*)(ws + OFF_BOXES);
  float* areas = (float*)(ws + OFF_AREAS);

  zero_kernel<<<dim3((BATCH * NBINS + 255) / 256), dim3(256), 0, stream>>>(hist, cnt);
  hist_kernel<<<dim3(NANCH / 256, BATCH), dim3(256), 0, stream>>>(rpn_probs, hist);
  scan_kernel<<<dim3(1), dim3(BATCH), 0, stream>>>(hist, tbin);
  compact_kernel<<<dim3(NANCH / 256, BATCH), dim3(256), 0, stream>>>(rpn_probs, tbin, cnt, keys);
  sort_kernel<<<dim3(BATCH), dim3(1024), 0, stream>>>(keys, cnt);
  gather_kernel<<<dim3((KTOP + 255) / 256, BATCH), dim3(256), 0, stream>>>(rpn_bbox, anchors, keys, boxes, areas);
  nms_kernel<<<dim3(BATCH), dim3(1024), 0, stream>>>(boxes, areas, out);
}